// MatchesLayerDistillationSegmentorV5_84361747628534
// MI455X (gfx1250) — compile-verified
//
#include <hip/hip_runtime.h>

// MI455X / gfx1250, wave32. f32 WMMA 16x16x4 for (a) seg-head GEMMs and
// (b) the 1-NN cross-term GEMM. Everything fits in L2; compute-bound on NN.
// f32 (not f16/bf16) WMMA keeps argmin + loss numerics identical to reference.

typedef __attribute__((ext_vector_type(2))) float v2f;
typedef __attribute__((ext_vector_type(8))) float v8f;

__device__ __forceinline__ v8f wmma_f32_4(v2f a, v2f b, v8f c) {
  // 8 args: (neg_a, A, neg_b, B, c_mod, C, reuse_a, reuse_b)
  return __builtin_amdgcn_wmma_f32_16x16x4_f32(false, a, false, b, (short)0, c,
                                               false, false);
}

// ---------------------------------------------------------------------------
// Kernel 1: logits = X[N,D] @ W[C,D]^T + bias[C], stored with row stride 32.
// One wave computes a 16-row x 32-col (2 tiles) output block; K in steps of 4.
// A layout: lanes 0-15 -> M=lane, K={k,k+1}; lanes 16-31 -> M=lane-16, K={k+2,k+3}
// B layout: lanes 0-15 -> N=lane, K={k,k+1}; lanes 16-31 -> N=lane-16, K={k+2,k+3}
// D layout: lane holds N=lane&15; VGPR v holds M = v + 8*(lane>=16)
// Out-of-range classes (n1 >= C) are handled branchlessly: load a clamped
// (valid) row and multiply by a 0/1 mask, so no per-iteration EXEC traffic
// and EXEC is all-ones at every WMMA.
// ---------------------------------------------------------------------------
__global__ void __launch_bounds__(128) seg_head_gemm(
    const float* __restrict__ X, const float* __restrict__ W,
    const float* __restrict__ bias, float* __restrict__ out,
    int N, int D, int C) {
  const int lane = threadIdx.x & 31;
  const int lm   = lane & 15;
  const int hf   = lane >> 4;
  const int wave = blockIdx.x * (blockDim.x >> 5) + (threadIdx.x >> 5);
  const int row_base = wave * 16;
  if (row_base >= N) return;  // wave-uniform

  const int n0 = lm;       // always < 22
  const int n1 = 16 + lm;  // valid only if < C
  const float mask1 = (n1 < C) ? 1.f : 0.f;
  const int n1c = (n1 < C) ? n1 : 0;  // clamped: always a valid row to load

  const float* xrow = X + (size_t)(row_base + lm) * D + 2 * hf;
  const float* w0   = W + (size_t)n0 * D + 2 * hf;
  const float* w1   = W + (size_t)n1c * D + 2 * hf;

  v8f acc0 = {}; v8f acc1 = {};
  for (int k = 0; k < D; k += 4) {
    v2f a;  a[0]  = xrow[k];        a[1]  = xrow[k + 1];
    v2f b0; b0[0] = w0[k];          b0[1] = w0[k + 1];
    v2f b1; b1[0] = w1[k] * mask1;  b1[1] = w1[k + 1] * mask1;
    acc0 = wmma_f32_4(a, b0, acc0);
    acc1 = wmma_f32_4(a, b1, acc1);
  }
  const float bb0 = bias[n0];
  const float bb1 = bias[n1c] * mask1;
#pragma unroll
  for (int k = 0; k < 8; ++k) {
    const int row = row_base + k + 8 * hf;
    out[(size_t)row * 32 + n0] = acc0[k] + bb0;
    out[(size_t)row * 32 + n1] = acc1[k] + bb1;
  }
}

// ---------------------------------------------------------------------------
// Kernel 2: brute-force 1-NN argmin.
// argmin_j (ss_i - 2 s_i.t_j + tt_j) == argmin_j (tt_j - 2 s_i.t_j).
// One wave owns 32 student rows (two A tiles); loops over teacher 16-col tiles.
// Per tile: one B load + tt, two f32 WMMAs (K=4, z-padded), 16 compare/selects.
// Lane's D column is fixed (N = lane&15), so per-lane running (min,idx) covers
// one column-residue class; final 4-step shfl_xor min-reduce per 16-lane half.
// All lane-half packing is branchless (v_cndmask), no EXEC divergence.
// ---------------------------------------------------------------------------
__global__ void __launch_bounds__(128) nn_argmin(
    const float* __restrict__ Sc, const float* __restrict__ Tc,
    int* __restrict__ idxOut, int Ns, int Nt) {
  const int lane = threadIdx.x & 31;
  const int lm   = lane & 15;
  const bool hi  = (lane >> 4) != 0;
  const int hf   = hi ? 1 : 0;
  const int wave = blockIdx.x * (blockDim.x >> 5) + (threadIdx.x >> 5);
  const int base = wave * 32;
  if (base >= Ns) return;  // wave-uniform

  const float* p0 = Sc + (size_t)3 * (base + lm);
  const float* p1 = Sc + (size_t)3 * (base + 16 + lm);
  const float s0x = p0[0], s0y = p0[1], s0z = p0[2];
  const float s1x = p1[0], s1y = p1[1], s1z = p1[2];
  v2f a0, a1;
  a0[0] = hi ? s0z : s0x;  a0[1] = hi ? 0.f : s0y;
  a1[0] = hi ? s1z : s1x;  a1[1] = hi ? 0.f : s1y;

  float bv0[8], bv1[8]; int bi0[8], bi1[8];
#pragma unroll
  for (int k = 0; k < 8; ++k) {
    bv0[k] = 3.4e38f; bv1[k] = 3.4e38f; bi0[k] = 0; bi1[k] = 0;
  }

  for (int jt = 0; jt < Nt; jt += 16) {
    const int t = jt + lm;
    const float* q = Tc + (size_t)3 * t;
    // Prefetch next tile's coords (global_prefetch_b8; speculative OOB on the
    // last iteration is silently dropped per ISA).
    __builtin_prefetch(q + 48, 0, 1);
    const float tx = q[0], ty = q[1], tz = q[2];
    v2f b;
    b[0] = hi ? tz : tx;
    b[1] = hi ? 0.f : ty;
    const float tt = tx * tx + ty * ty + tz * tz;
    v8f c0 = {}; v8f c1 = {};
    c0 = wmma_f32_4(a0, b, c0);
    c1 = wmma_f32_4(a1, b, c1);
#pragma unroll
    for (int k = 0; k < 8; ++k) {
      const float v0 = fmaf(-2.f, c0[k], tt);
      if (v0 < bv0[k]) { bv0[k] = v0; bi0[k] = t; }
      const float v1 = fmaf(-2.f, c1[k], tt);
      if (v1 < bv1[k]) { bv1[k] = v1; bi1[k] = t; }
    }
  }

#pragma unroll
  for (int k = 0; k < 8; ++k) {
    float v = bv0[k]; int ix = bi0[k];
#pragma unroll
    for (int m = 1; m < 16; m <<= 1) {
      const float ov = __shfl_xor(v, m, 32);
      const int   oi = __shfl_xor(ix, m, 32);
      if (ov < v || (ov == v && oi < ix)) { v = ov; ix = oi; }
    }
    if (lm == 0) idxOut[base + k + 8 * hf] = ix;

    v = bv1[k]; ix = bi1[k];
#pragma unroll
    for (int m = 1; m < 16; m <<= 1) {
      const float ov = __shfl_xor(v, m, 32);
      const int   oi = __shfl_xor(ix, m, 32);
      if (ov < v || (ov == v && oi < ix)) { v = ov; ix = oi; }
    }
    if (lm == 0) idxOut[base + 16 + k + 8 * hf] = ix;
  }
}

// ---------------------------------------------------------------------------
// Kernel 3: per-student KL (T=2) + CE contributions; deterministic per-block
// partial sums (no atomics — graph-replay safe, ws rewritten every launch).
// ---------------------------------------------------------------------------
__global__ void __launch_bounds__(256) loss_partials(
    const float* __restrict__ sLog, const float* __restrict__ tLog,
    const int* __restrict__ nnIdx, const int* __restrict__ segment,
    float* __restrict__ partials, int Ns, int C) {
  __shared__ float sCe[8];
  __shared__ float sKl[8];
  const int i = blockIdx.x * blockDim.x + threadIdx.x;
  float ce = 0.f, kl = 0.f;
  if (i < Ns) {
    const float* s = sLog + (size_t)i * 32;
    const float* m = tLog + (size_t)nnIdx[i] * 32;
    float smax = -3.4e38f, mmax = -3.4e38f;
    for (int c = 0; c < C; ++c) {
      smax = fmaxf(smax, s[c]);
      mmax = fmaxf(mmax, m[c]);
    }
    float ses = 0.f, ses2 = 0.f, mes = 0.f;
    for (int c = 0; c < C; ++c) {
      ses  += expf(s[c] - smax);                 // for CE log-softmax
      ses2 += expf((s[c] - smax) * 0.5f);        // student logits / T
      mes  += expf((m[c] - mmax) * 0.5f);        // teacher logits / T
    }
    const float lses2 = logf(ses2);
    const float lmes  = logf(mes);
    for (int c = 0; c < C; ++c) {
      const float lp  = (m[c] - mmax) * 0.5f - lmes;   // log teacher_p
      const float pc  = expf(lp);
      const float slp = (s[c] - smax) * 0.5f - lses2;  // student log-softmax(/T)
      kl += pc * (lp - slp);
    }
    ce = (smax + logf(ses)) - s[segment[i]];
  }
#pragma unroll
  for (int m = 16; m >= 1; m >>= 1) {
    ce += __shfl_xor(ce, m, 32);
    kl += __shfl_xor(kl, m, 32);
  }
  const int wid = threadIdx.x >> 5;
  if ((threadIdx.x & 31) == 0) { sCe[wid] = ce; sKl[wid] = kl; }
  __syncthreads();
  if (threadIdx.x == 0) {
    float tc = 0.f, tk = 0.f;
    const int nw = blockDim.x >> 5;
    for (int w = 0; w < nw; ++w) { tc += sCe[w]; tk += sKl[w]; }
    partials[2 * blockIdx.x]     = tc;
    partials[2 * blockIdx.x + 1] = tk;
  }
}

__global__ void finalize_loss(const float* __restrict__ partials, int nblk,
                              float* __restrict__ out, int Ns) {
  if (threadIdx.x == 0 && blockIdx.x == 0) {
    float ce = 0.f, kl = 0.f;
    for (int b = 0; b < nblk; ++b) {
      ce += partials[2 * b];
      kl += partials[2 * b + 1];
    }
    const float seg_loss = ce / (float)Ns;
    const float kl_loss  = 0.2f * (kl / (float)Ns * 4.0f);  // * T^2, T=2
    out[0] = seg_loss + kl_loss;
    out[1] = seg_loss;
    out[2] = kl_loss;
  }
}

extern "C" void kernel_launch(void* const* d_in, const int* in_sizes, int n_in,
                              void* d_out, int out_size, void* d_ws,
                              size_t ws_size, hipStream_t stream) {
  (void)n_in; (void)out_size; (void)ws_size;
  const float* s_feat   = (const float*)d_in[0];
  const float* t_feat   = (const float*)d_in[1];
  const float* s_coords = (const float*)d_in[2];
  const float* t_coords = (const float*)d_in[3];
  const float* seg_W    = (const float*)d_in[4];
  const float* seg_b    = (const float*)d_in[5];
  const float* seg_tW   = (const float*)d_in[6];
  const float* seg_tb   = (const float*)d_in[7];
  const int*   segment  = (const int*)d_in[8];

  const int Ns = in_sizes[2] / 3;        // 16384
  const int Nt = in_sizes[3] / 3;        // 16384
  const int C  = in_sizes[5];            // 22
  const int D  = in_sizes[0] / Ns;       // 64

  // Workspace layout (floats): t_logits[Nt*32] | s_logits[Ns*32] | idx[Ns] | partials
  float* tLog     = (float*)d_ws;
  float* sLog     = tLog + (size_t)Nt * 32;
  int*   nnIdx    = (int*)(sLog + (size_t)Ns * 32);
  float* partials = (float*)(nnIdx + Ns);

  const int WPB = 4;  // waves per block (blockDim 128 = 4 x wave32)

  const int wavesT = Nt / 16;
  seg_head_gemm<<<(wavesT + WPB - 1) / WPB, 128, 0, stream>>>(
      t_feat, seg_tW, seg_tb, tLog, Nt, D, C);

  const int wavesS = Ns / 16;
  seg_head_gemm<<<(wavesS + WPB - 1) / WPB, 128, 0, stream>>>(
      s_feat, seg_W, seg_b, sLog, Ns, D, C);

  const int wavesNN = Ns / 32;
  nn_argmin<<<(wavesNN + WPB - 1) / WPB, 128, 0, stream>>>(
      s_coords, t_coords, nnIdx, Ns, Nt);

  const int nblk = (Ns + 255) / 256;
  loss_partials<<<nblk, 256, 0, stream>>>(sLog, tLog, nnIdx, segment, partials,
                                          Ns, C);
  finalize_loss<<<1, 32, 0, stream>>>(partials, nblk, (float*)d_out, Ns);
}